// MultiboxLoss_84490596647030
// MI455X (gfx1250) — compile-verified
//
#include <hip/hip_runtime.h>

#define TK   256   // predictions per image
#define KMAX 32    // max gt objects per image
#define WAVE 32

typedef float v2f __attribute__((ext_vector_type(2)));
typedef float v8f __attribute__((ext_vector_type(8)));

// ---------- gfx1250 async global->LDS helpers ----------
__device__ __forceinline__ unsigned lds_addr_of(const void* p) {
  // generic -> LDS(addrspace 3) -> byte offset within the workgroup LDS segment
  return (unsigned)(unsigned long long)(const __attribute__((address_space(3))) void*)p;
}

__device__ __forceinline__ void async_copy_b32(void* lds_dst, const void* gsrc) {
  asm volatile("global_load_async_to_lds_b32 %0, %1, off"
               :: "v"(lds_addr_of(lds_dst)), "v"(gsrc) : "memory");
}

__device__ __forceinline__ void async_copy_b128(void* lds_dst, const void* gsrc) {
  asm volatile("global_load_async_to_lds_b128 %0, %1, off"
               :: "v"(lds_addr_of(lds_dst)), "v"(gsrc) : "memory");
}

__device__ __forceinline__ void wait_async0() {
#if __has_builtin(__builtin_amdgcn_s_wait_asynccnt)
  __builtin_amdgcn_s_wait_asynccnt(0);
#else
  asm volatile("s_wait_asynccnt 0x0" ::: "memory");
#endif
}

// One wave32 workgroup per batch image. Runs the Jonker-Volgenant Hungarian
// solve with the cost matrix and all solver state resident in LDS; the inner
// column scans (256 cols) are parallelized 8 cols/lane with shfl-xor argmin.
__global__ __launch_bounds__(WAVE) void hungarian_loss_kernel(
    const float* __restrict__ conf,   // (B, 256, 1)
    const float* __restrict__ ploc,   // (B, 256, 4)
    const float* __restrict__ glab,   // (B, 32, 1)
    const float* __restrict__ gloc,   // (B, 32, 4)
    const int*   __restrict__ nobj,   // (B, 1)
    float*       __restrict__ partial // (B) per-image sum of smoothL1 dst_pp
) {
  const int b    = blockIdx.x;
  const int lane = threadIdx.x;

  __shared__ __align__(16) float sConf[TK];
  __shared__ __align__(16) float sPloc[TK * 4];
  __shared__ __align__(16) float sGlab[KMAX];
  __shared__ __align__(16) float sGloc[KMAX * 4];
  __shared__ float sCost[KMAX][TK];
  __shared__ float sU[KMAX + 1];
  __shared__ float sV[TK + 1];
  __shared__ float sMinv[TK + 1];
  __shared__ int   sWay[TK + 1];
  __shared__ int   sP[TK + 1];
  __shared__ int   sUsed[TK + 1];
  __shared__ int   sPairR[KMAX];
  __shared__ int   sPairT[KMAX];
  __shared__ int   sCount;

  const float* gC = conf + (size_t)b * TK;
  const float* gP = ploc + (size_t)b * TK * 4;
  const float* gL = glab + (size_t)b * KMAX;
  const float* gG = gloc + (size_t)b * KMAX * 4;

  // ---- stage inputs to LDS via gfx1250 async copies (TENSOR/ASYNC path) ----
  for (int i = lane; i < TK; i += WAVE)  async_copy_b32(&sConf[i], gC + i);
  for (int i = lane; i < TK; i += WAVE)  async_copy_b128(&sPloc[i * 4], gP + i * 4);
  async_copy_b32(&sGlab[lane], gL + lane);          // KMAX == WAVE
  async_copy_b128(&sGloc[lane * 4], gG + lane * 4); // KMAX == WAVE
  wait_async0();
  __syncthreads();

  int n = nobj[b];
  n = n < 0 ? 0 : (n > KMAX ? KMAX : n);

  // ---- build cost matrix: 1.0*|dlabel| + 1.5*L1(dloc), rows=gt, cols=pred ----
  for (int r = 0; r < n; ++r) {
    const float g  = sGlab[r];
    const float g0 = sGloc[r * 4 + 0], g1 = sGloc[r * 4 + 1];
    const float g2 = sGloc[r * 4 + 2], g3 = sGloc[r * 4 + 3];
    for (int t = lane; t < TK; t += WAVE) {
      float c = fabsf(g - sConf[t]);
      float d = fabsf(g0 - sPloc[t * 4 + 0]) + fabsf(g1 - sPloc[t * 4 + 1]) +
                fabsf(g2 - sPloc[t * 4 + 2]) + fabsf(g3 - sPloc[t * 4 + 3]);
      sCost[r][t] = c + 1.5f * d;
    }
  }
  for (int j = lane; j <= TK; j += WAVE) { sV[j] = 0.f; sP[j] = 0; sWay[j] = 0; }
  if (lane <= KMAX) sU[lane] = 0.f;
  __syncthreads();

  const float INFV = __builtin_inff();

  // ---- Jonker-Volgenant ----
  for (int i = 1; i <= n; ++i) {
    if (lane == 0) sP[0] = i;
    for (int j = lane; j <= TK; j += WAVE) { sMinv[j] = INFV; sUsed[j] = 0; }
    __syncthreads();
    int j0 = 0;
    while (true) {
      if (lane == 0) sUsed[j0] = 1;
      __syncthreads();
      const int   i0  = sP[j0];
      const float ui0 = sU[i0];
      float best = INFV; int bestj = TK + 1;
      #pragma unroll
      for (int s = 0; s < TK / WAVE; ++s) {
        const int j = 1 + lane + s * WAVE;     // j in 1..256, ascending per lane
        if (!sUsed[j]) {
          float cur = sCost[i0 - 1][j - 1] - ui0 - sV[j];
          float mv  = sMinv[j];
          if (cur < mv) { mv = cur; sMinv[j] = cur; sWay[j] = j0; }
          if (mv < best) { best = mv; bestj = j; }
        }
      }
      // wave32 argmin; smallest column index wins ties (matches np.argmin)
      #pragma unroll
      for (int off = 16; off > 0; off >>= 1) {
        float ob = __shfl_xor(best, off, WAVE);
        int   oj = __shfl_xor(bestj, off, WAVE);
        if (ob < best || (ob == best && oj < bestj)) { best = ob; bestj = oj; }
      }
      const float delta = best;
      const int   j1    = bestj;
      __syncthreads();
      #pragma unroll
      for (int s = 0; s < TK / WAVE; ++s) {
        const int j = 1 + lane + s * WAVE;
        if (sUsed[j]) { sU[sP[j]] += delta; sV[j] -= delta; }
        else          { sMinv[j] -= delta; }
      }
      if (lane == 0) sU[sP[0]] += delta;  // virtual column 0 (always used)
      __syncthreads();
      j0 = j1;
      if (sP[j0] == 0) break;
    }
    // augment along the alternating path (uniform across the wave)
    while (j0 != 0) {
      const int j1a = sWay[j0];
      if (lane == 0) sP[j0] = sP[j1a];
      __syncthreads();
      j0 = j1a;
    }
  }

  // ---- extract matched (gt r, pred t) pairs ----
  if (lane == 0) {
    int cnt = 0;
    for (int j = 1; j <= TK; ++j) {
      const int pr = sP[j];
      if (pr > 0 && cnt < KMAX) { sPairR[cnt] = pr - 1; sPairT[cnt] = j - 1; ++cnt; }
    }
    sCount = cnt;
  }
  __syncthreads();
  const int cnt = sCount;

  // ---- loss: cls term is exactly 0 (log_softmax over size-1 axis);
  //      smooth-L1 location term reduced on the matrix unit:
  //      D = A(16x4 smoothL1 values) x B(4x16 ones) => per-pair row sums. ----
  float total = 0.f;
#if __has_builtin(__builtin_amdgcn_wmma_f32_16x16x4_f32)
  {
    v8f acc = {0.f, 0.f, 0.f, 0.f, 0.f, 0.f, 0.f, 0.f};
    const int half = lane >> 4;     // A layout: lanes 0-15 hold K=0,1; 16-31 hold K=2,3
    const int m    = lane & 15;
    const int k0   = half * 2;
    v2f Bv; Bv.x = 1.0f; Bv.y = 1.0f;
    #pragma unroll
    for (int g = 0; g < 2; ++g) {   // two groups of 16 pairs cover n <= 32
      const int q = g * 16 + m;
      float s0 = 0.f, s1 = 0.f;
      if (q < cnt) {
        const int r = sPairR[q], t = sPairT[q];
        float d0 = sPloc[t * 4 + k0]     - sGloc[r * 4 + k0];
        float d1 = sPloc[t * 4 + k0 + 1] - sGloc[r * 4 + k0 + 1];
        float a0 = fabsf(d0), a1 = fabsf(d1);
        s0 = (a0 < 1.f) ? 0.5f * a0 * a0 : a0 - 0.5f;
        s1 = (a1 < 1.f) ? 0.5f * a1 * a1 : a1 - 0.5f;
      }
      v2f Av; Av.x = s0; Av.y = s1;
      acc = __builtin_amdgcn_wmma_f32_16x16x4_f32(
          /*neg_a=*/false, Av, /*neg_b=*/false, Bv,
          /*c_mod=*/(short)0, acc, /*reuse_a=*/false, /*reuse_b=*/false);
    }
    // D column N=0 lives in lane 0 (M=0..7) and lane 16 (M=8..15)
    float ls = acc[0] + acc[1] + acc[2] + acc[3] + acc[4] + acc[5] + acc[6] + acc[7];
    total = __shfl(ls, 0, WAVE) + __shfl(ls, 16, WAVE);
  }
#else
  {
    float ls = 0.f;
    for (int q = lane; q < cnt; q += WAVE) {
      const int r = sPairR[q], t = sPairT[q];
      #pragma unroll
      for (int k = 0; k < 4; ++k) {
        float d = sPloc[t * 4 + k] - sGloc[r * 4 + k];
        float a = fabsf(d);
        ls += (a < 1.f) ? 0.5f * a * a : a - 0.5f;
      }
    }
    #pragma unroll
    for (int off = 16; off > 0; off >>= 1) ls += __shfl_xor(ls, off, WAVE);
    total = ls;
  }
#endif

  if (lane == 0) partial[b] = total;
}

__global__ void finalize_kernel(const float* __restrict__ partial,
                                float* __restrict__ out, int B) {
  if (threadIdx.x == 0 && blockIdx.x == 0) {
    float s = 0.f;
    for (int i = 0; i < B; ++i) s += partial[i];
    // loss = CLS*0 + DST*dst_loss + MASK*0 ; dst_loss = sum/B
    out[0] = 1.5f * s / (float)B;
  }
}

extern "C" void kernel_launch(void* const* d_in, const int* in_sizes, int n_in,
                              void* d_out, int out_size, void* d_ws, size_t ws_size,
                              hipStream_t stream) {
  const float* conf = (const float*)d_in[0];  // confidence
  const float* ploc = (const float*)d_in[1];  // predicted_locations
  // d_in[2] predicted_masks: unused by the loss (mask_loss == 0) -> never read
  const float* glab = (const float*)d_in[3];  // gt_labels
  const float* gloc = (const float*)d_in[4];  // gt_locations
  // d_in[5] gt_masks: unused
  const int*   nobj = (const int*)d_in[6];    // num_objects

  const int B = in_sizes[6];                  // (B,1) -> B elements
  float* partial = (float*)d_ws;

  hungarian_loss_kernel<<<B, WAVE, 0, stream>>>(conf, ploc, glab, gloc, nobj, partial);
  finalize_kernel<<<1, 32, 0, stream>>>(partial, (float*)d_out, B);
}